// AttnDecoderRnn_82257213653299
// MI455X (gfx1250) — compile-verified
//
#include <hip/hip_runtime.h>

// ---------------------------------------------------------------------------
// AttnDecoderRNN (Bahdanau attention + GRU + vocab projection) for gfx1250.
// GEMMs: V_WMMA_F32_16X16X4_F32 with double-buffered async global->LDS
// staging (GLOBAL_LOAD_ASYNC_TO_LDS_B128, ASYNCcnt) shared by all 8 waves.
// ---------------------------------------------------------------------------

typedef __attribute__((ext_vector_type(2))) float v2f;
typedef __attribute__((ext_vector_type(8))) float v8f;

#define HID   1024
#define VOC   32000
#define BATCH 64
#define SLEN  50
#define TSTEP 25
#define SOSTK 1

// LDS row padding: 20 floats (80 B) per 16-float row.
//  - 16B-aligned b128 async stores for kk in {0,4,8,12}
//  - conflict-free b64 fragment reads (row*20 mod 64 distinct over 16 rows)
#define LPAD 20

// Async copy of 16 bytes global -> LDS (per lane), tracked by ASYNCcnt.
__device__ __forceinline__ void async_copy_b128(unsigned ldsByteAddr,
                                                const float* base,
                                                unsigned gByteOff)
{
    asm volatile("global_load_async_to_lds_b128 %0, %1, %2"
                 :: "v"(ldsByteAddr), "v"(gByteOff), "s"(base)
                 : "memory");
}

__device__ __forceinline__ unsigned lds_addr(const void* p)
{
    // LDS aperture: flat address low 32 bits are the LDS byte offset.
    return (unsigned)(unsigned long long)p;
}

// C[M,N] = A[M,K] @ B[N,K]^T + bias[N]
// Block: 256 threads = 8 waves as 4(M) x 2(N); wave tile 16x64;
// block tile 64(M) x 128(N); K consumed in 16-wide chunks, double buffered.
// Requires M%64==0, N%128==0, K%16==0 (true for all GEMMs in this model);
// grids are exact so no tile guards (barriers require full participation).
__global__ __launch_bounds__(256)
void gemm_atb_wmma(const float* __restrict__ A, const float* __restrict__ Bm,
                   const float* __restrict__ bias, float* __restrict__ C,
                   int M, int N, int K, long ldc)
{
    __shared__ float As[2][64][LPAD];    // A chunk: 64 rows x 16 K
    __shared__ float Bs[2][128][LPAD];   // B chunk: 128 rows x 16 K

    const int tid  = threadIdx.x;
    const int lane = tid & 31;
    const int wave = tid >> 5;
    const int wm   = wave >> 1;                  // 0..3  (M tile in block)
    const int wn   = wave & 1;                   // 0..1  (N half in block)
    const int mBlock = blockIdx.y * 64;
    const int nBlock = blockIdx.x * 128;

    const int lh = lane >> 4;                    // 0 or 1
    const int lm = lane & 15;

    // Per-thread async-staging assignment: 3 x b128 per thread per chunk
    // (A: 64x16 floats = 256 x float4; B: 128x16 = 512 x float4).
    const int srow = tid >> 2;                   // 0..63
    const int skk  = (tid & 3) << 2;             // 0,4,8,12

    const long aRowOff = ((long)(mBlock + srow) * K + skk) * 4;
    const long bRowOff0 = ((long)(nBlock + srow) * K + skk) * 4;
    const long bRowOff1 = ((long)(nBlock + srow + 64) * K + skk) * 4;

    v8f acc[4] = {};

    const int nCh = K >> 4;

    // Prologue: stage chunk 0 into buffer 0.
    {
        const unsigned kb = 0;
        async_copy_b128(lds_addr(&As[0][srow][skk]), A,  (unsigned)(aRowOff  + kb));
        async_copy_b128(lds_addr(&Bs[0][srow][skk]), Bm, (unsigned)(bRowOff0 + kb));
        async_copy_b128(lds_addr(&Bs[0][srow + 64][skk]), Bm, (unsigned)(bRowOff1 + kb));
    }

    for (int c = 0; c < nCh; ++c) {
        const int buf = c & 1;
        if (c + 1 < nCh) {
            const int nb = (c + 1) & 1;
            const unsigned kb = (unsigned)((c + 1) << 6);   // 16 floats = 64 bytes
            async_copy_b128(lds_addr(&As[nb][srow][skk]), A,  (unsigned)(aRowOff  + kb));
            async_copy_b128(lds_addr(&Bs[nb][srow][skk]), Bm, (unsigned)(bRowOff0 + kb));
            async_copy_b128(lds_addr(&Bs[nb][srow + 64][skk]), Bm, (unsigned)(bRowOff1 + kb));
            asm volatile("s_wait_asynccnt 0x3" ::: "memory");  // chunk c landed
        } else {
            asm volatile("s_wait_asynccnt 0x0" ::: "memory");
        }
        __syncthreads();

        // Bulk-load all fragments of this chunk from LDS, then 16 WMMAs.
        v2f a[4];
        v2f bf[4][4];
#pragma unroll
        for (int ks = 0; ks < 4; ++ks)
            a[ks] = *(const v2f*)&As[buf][wm * 16 + lm][4 * ks + 2 * lh];
#pragma unroll
        for (int ks = 0; ks < 4; ++ks)
#pragma unroll
            for (int j = 0; j < 4; ++j)
                bf[ks][j] = *(const v2f*)&Bs[buf][wn * 64 + j * 16 + lm][4 * ks + 2 * lh];

#pragma unroll
        for (int ks = 0; ks < 4; ++ks) {
#pragma unroll
            for (int j = 0; j < 4; ++j)
                acc[j] = __builtin_amdgcn_wmma_f32_16x16x4_f32(
                    false, a[ks], false, bf[ks][j], (short)0, acc[j], false, false);
        }
        __syncthreads();   // everyone done reading buf before it is re-staged
    }

    // Epilogue: C/D layout -> row = mBase + v + 8*lh, col = nBase + j*16 + lm
    const int row0 = mBlock + wm * 16 + 8 * lh;
#pragma unroll
    for (int j = 0; j < 4; ++j) {
        const int col = nBlock + wn * 64 + j * 16 + lm;
        const float bb = bias[col];
#pragma unroll
        for (int v = 0; v < 8; ++v) {
            C[(long)(row0 + v) * ldc + col] = acc[j][v] + bb;
        }
    }
}

// Attention scores + softmax + context, plus embedding gather into xcat.
__global__ __launch_bounds__(256)
void attn_context(const float* __restrict__ q,      // [B,H]   (h @ Wa^T + ba)
                  const float* __restrict__ keys,   // [B,S,H] (Ua projection)
                  const float* __restrict__ enc,    // [B,S,H] encoder_outputs
                  const float* __restrict__ Va,     // [H]
                  const float* __restrict__ bv,     // [1]
                  const float* __restrict__ emb,    // [V,H]
                  const int*   __restrict__ target, // [B,T]
                  float* __restrict__ xcat,         // [B,2H]  [emb | context]
                  int t)
{
    __shared__ float w[SLEN];
    const int b    = blockIdx.x;
    const int lane = threadIdx.x & 31;
    const int wave = threadIdx.x >> 5;

    const float* qb = q + (long)b * HID;
    for (int s = wave; s < SLEN; s += 8) {
        const float* kp = keys + ((long)b * SLEN + s) * HID;
        float acc = 0.0f;
        for (int h = lane; h < HID; h += 32)
            acc += tanhf(qb[h] + kp[h]) * Va[h];
#pragma unroll
        for (int off = 16; off > 0; off >>= 1) acc += __shfl_xor(acc, off, 32);
        if (lane == 0) w[s] = acc + bv[0];
    }
    __syncthreads();

    if (wave == 0) {
        float m = -3.402823466e38f;
        for (int s = lane; s < SLEN; s += 32) m = fmaxf(m, w[s]);
#pragma unroll
        for (int off = 16; off > 0; off >>= 1) m = fmaxf(m, __shfl_xor(m, off, 32));
        float sum = 0.0f;
        for (int s = lane; s < SLEN; s += 32) {
            float e = __expf(w[s] - m);
            w[s] = e;
            sum += e;
        }
#pragma unroll
        for (int off = 16; off > 0; off >>= 1) sum += __shfl_xor(sum, off, 32);
        const float inv = 1.0f / sum;
        for (int s = lane; s < SLEN; s += 32) w[s] *= inv;
    }
    __syncthreads();

    const int tok = (t == 0) ? SOSTK : target[b * TSTEP + (t - 1)];
    const float* eb   = emb + (long)tok * HID;
    const float* encb = enc + (long)b * SLEN * HID;
    float* xb = xcat + (long)b * (2 * HID);
    for (int h = threadIdx.x; h < HID; h += 256) {
        float c = 0.0f;
#pragma unroll 5
        for (int s = 0; s < SLEN; ++s) c += w[s] * encb[(long)s * HID + h];
        xb[h]       = eb[h];
        xb[HID + h] = c;
    }
}

// GRU gate fusion (PyTorch gate order r,z,n): h' = (1-z)*n + z*h
__global__ __launch_bounds__(256)
void gru_gate(const float* __restrict__ gx,  // [B,3H]
              const float* __restrict__ gh,  // [B,3H]
              const float* __restrict__ h,   // [B,H]
              float* __restrict__ hnext)     // [B,H]
{
    const int i  = blockIdx.x * blockDim.x + threadIdx.x;   // 0 .. B*H-1
    const int b  = i >> 10;
    const int hh = i & (HID - 1);
    const float* gxb = gx + (long)b * (3 * HID);
    const float* ghb = gh + (long)b * (3 * HID);
    const float rx = gxb[hh], zx = gxb[HID + hh], nx = gxb[2 * HID + hh];
    const float rh = ghb[hh], zh = ghb[HID + hh], nh = ghb[2 * HID + hh];
    const float r = 1.0f / (1.0f + __expf(-(rx + rh)));
    const float z = 1.0f / (1.0f + __expf(-(zx + zh)));
    const float n = tanhf(nx + r * nh);
    hnext[i] = (1.0f - z) * n + z * h[i];
}

extern "C" void kernel_launch(void* const* d_in, const int* in_sizes, int n_in,
                              void* d_out, int out_size, void* d_ws, size_t ws_size,
                              hipStream_t stream)
{
    (void)in_sizes; (void)n_in; (void)out_size; (void)ws_size;

    const float* enc_out = (const float*)d_in[0];   // [B,S,H]
    const float* enc_hid = (const float*)d_in[1];   // [1,B,H] -> h0
    const int*   target  = (const int*)  d_in[2];   // [B,T]
    const float* emb     = (const float*)d_in[3];   // [V,H]
    const float* Wa      = (const float*)d_in[4];   // [H,H]
    const float* ba      = (const float*)d_in[5];   // [H]
    const float* Ua      = (const float*)d_in[6];   // [H,H]
    const float* bu      = (const float*)d_in[7];   // [H]
    const float* Va      = (const float*)d_in[8];   // [H,1]
    const float* bvv     = (const float*)d_in[9];   // [1]
    const float* W_ih    = (const float*)d_in[10];  // [3H,2H]
    const float* b_ih    = (const float*)d_in[11];  // [3H]
    const float* W_hh    = (const float*)d_in[12];  // [3H,H]
    const float* b_hh    = (const float*)d_in[13];  // [3H]
    const float* Wout    = (const float*)d_in[14];  // [V,H]
    const float* bout    = (const float*)d_in[15];  // [V]
    float* out = (float*)d_out;                     // [B,T,V]

    float* ws   = (float*)d_ws;
    float* keys = ws;                                  // B*S*H
    float* q    = keys + (long)BATCH * SLEN * HID;     // B*H
    float* xcat = q    + (long)BATCH * HID;            // B*2H
    float* gx   = xcat + (long)BATCH * 2 * HID;        // B*3H
    float* gh   = gx   + (long)BATCH * 3 * HID;        // B*3H
    float* h0b  = gh   + (long)BATCH * 3 * HID;        // B*H
    float* h1b  = h0b  + (long)BATCH * HID;            // B*H

    const dim3 blk(256);

    // keys_proj = encoder_outputs @ Ua^T + bu   (M=3200, N=1024, K=1024)
    gemm_atb_wmma<<<dim3(HID / 128, (BATCH * SLEN) / 64), blk, 0, stream>>>(
        enc_out, Ua, bu, keys, BATCH * SLEN, HID, HID, (long)HID);

    for (int t = 0; t < TSTEP; ++t) {
        const float* hcur = (t == 0) ? enc_hid : (((t - 1) & 1) ? h1b : h0b);
        float*       hnxt = (t & 1) ? h1b : h0b;

        // q = h @ Wa^T + ba                        (M=64, N=1024, K=1024)
        gemm_atb_wmma<<<dim3(HID / 128, 1), blk, 0, stream>>>(
            hcur, Wa, ba, q, BATCH, HID, HID, (long)HID);

        attn_context<<<dim3(BATCH), blk, 0, stream>>>(
            q, keys, enc_out, Va, bvv, emb, target, xcat, t);

        // gx = xcat @ W_ih^T + b_ih                (M=64, N=3072, K=2048)
        gemm_atb_wmma<<<dim3((3 * HID) / 128, 1), blk, 0, stream>>>(
            xcat, W_ih, b_ih, gx, BATCH, 3 * HID, 2 * HID, (long)(3 * HID));

        // gh = h @ W_hh^T + b_hh                   (M=64, N=3072, K=1024)
        gemm_atb_wmma<<<dim3((3 * HID) / 128, 1), blk, 0, stream>>>(
            hcur, W_hh, b_hh, gh, BATCH, 3 * HID, HID, (long)(3 * HID));

        gru_gate<<<dim3((BATCH * HID) / 256), blk, 0, stream>>>(gx, gh, hcur, hnxt);

        // logits_t = h_next @ Wout^T + bout -> out[:, t, :]
        // (M=64, N=32000, K=1024; Wout L2-resident across steps; B tile now
        //  staged once per block in LDS and shared by all 4 M-waves.)
        gemm_atb_wmma<<<dim3(VOC / 128, 1), blk, 0, stream>>>(
            hnxt, Wout, bout, out + (long)t * VOC, BATCH, VOC, HID, (long)TSTEP * VOC);
    }
}